// User2ItemLayer_7224134991886
// MI455X (gfx1250) — compile-verified
//
#include <hip/hip_runtime.h>
#include <math.h>

#define V_SZ 200000
#define D_SZ 64
#define B_SZ 4096
#define T_SZ 50
#define S_SZ 64

typedef __attribute__((ext_vector_type(2))) float v2f;
typedef __attribute__((ext_vector_type(8))) float v8f;

__device__ __forceinline__ float logq_of(int id) {
    float idf = (float)id;
    return logf((logf(idf + 2.0f) - logf(idf + 1.0f)) / logf((float)(V_SZ + 1)));
}

// ---------------------------------------------------------------------------
// Kernel 1: per-batch-row attention + softmax + pooling + output + ut_1
// One block per batch row b. 256 threads = 8 waves (wave32).
// ---------------------------------------------------------------------------
__global__ __launch_bounds__(256) void attn_pool_kernel(
    const float* __restrict__ item_table,
    const float* __restrict__ pos_table,
    const float* __restrict__ att_W,
    const float* __restrict__ att_b,
    const float* __restrict__ prelu_alpha,
    const int*   __restrict__ items_id,
    const int*   __restrict__ position_id,
    const int*   __restrict__ target_id,
    const int*   __restrict__ keys_length,
    float*       __restrict__ out,        // d_out[0..B)
    float*       __restrict__ ut1,        // ws: B x D
    int*         __restrict__ result_id)  // ws: B
{
    const int b    = blockIdx.x;
    const int tid  = threadIdx.x;
    const int lane = tid & 31;
    const int wave = tid >> 5;

    __shared__ float sItems[T_SZ][D_SZ];
    __shared__ float sW[2 * D_SZ];
    __shared__ float sTgt[D_SZ];
    __shared__ float sScore[T_SZ];
    __shared__ float sWeight[T_SZ];
    __shared__ float sNz[T_SZ];
    __shared__ float sRed[D_SZ];

    if (tid < 2 * D_SZ) sW[tid] = att_W[tid];
    if (tid < D_SZ) {
        int tg = target_id[b];
        sTgt[tid] = item_table[(long long)tg * D_SZ + tid];
    }
    __syncthreads();

    const float bb = att_b[0];

    // Gather item rows once into LDS; compute attention score per t on the fly.
    for (int t = wave; t < T_SZ; t += 8) {
        int iid = items_id[b * T_SZ + t];
        int pid = position_id[b * T_SZ + t];
        float f0 = item_table[(long long)iid * D_SZ + lane];
        float f1 = item_table[(long long)iid * D_SZ + lane + 32];
        float p0 = pos_table[pid * D_SZ + lane];
        float p1 = pos_table[pid * D_SZ + lane + 32];
        sItems[t][lane]      = f0;
        sItems[t][lane + 32] = f1;
        float partial = f0 * sW[lane] + f1 * sW[lane + 32]
                      + p0 * sW[64 + lane] + p1 * sW[96 + lane];
        #pragma unroll
        for (int off = 16; off > 0; off >>= 1)
            partial += __shfl_xor(partial, off, 32);
        if (lane == 0) {
            sScore[t] = tanhf(partial + bb);
            sNz[t]    = (iid != 0) ? 1.0f : 0.0f;
        }
    }
    __syncthreads();

    const int kl = keys_length[b];   // in [1, T]

    // Masked softmax over T, done by wave 0 (lane handles t=lane and t=lane+32).
    if (tid < 32) {
        int t1 = lane + 32;
        float s0 = (lane < kl) ? sScore[lane] : -INFINITY;
        float s1 = (t1 < T_SZ && t1 < kl) ? sScore[t1] : -INFINITY;
        float m = fmaxf(s0, s1);
        #pragma unroll
        for (int off = 16; off > 0; off >>= 1)
            m = fmaxf(m, __shfl_xor(m, off, 32));
        float e0 = (lane < kl) ? expf(s0 - m) : 0.0f;
        float e1 = (t1 < T_SZ && t1 < kl) ? expf(s1 - m) : 0.0f;
        float sum = e0 + e1;
        #pragma unroll
        for (int off = 16; off > 0; off >>= 1)
            sum += __shfl_xor(sum, off, 32);
        float inv = 1.0f / sum;
        sWeight[lane] = e0 * inv;
        if (t1 < T_SZ) sWeight[t1] = e1 * inv;
    }
    __syncthreads();

    // Pooling + PReLU + ut_1 (excludes last timestep, zero-id padded).
    if (tid < D_SZ) {
        const int d = tid;
        float sp = 0.0f, ut = 0.0f;
        for (int t = 0; t < T_SZ; ++t) {
            float wx = sWeight[t] * sItems[t][d];
            sp += wx;
            if (t < T_SZ - 1) ut += wx * sNz[t];
        }
        float a   = prelu_alpha[d];
        float spv = (sp > 0.0f) ? sp : a * sp;
        ut1[b * D_SZ + d] = ut;
        sRed[d] = spv * sTgt[d];
    }
    __syncthreads();

    if (tid == 0) {
        float acc = 0.0f;
        for (int d = 0; d < D_SZ; ++d) acc += sRed[d];   // fixed order: deterministic
        out[b] = acc;
        result_id[b] = items_id[b * T_SZ + (kl - 1)];
    }
}

// ---------------------------------------------------------------------------
// Kernel 2: samp_logits = ut_1 (B x 64) @ samp_w^T (64 x 64) via
// V_WMMA_F32_16X16X4_F32 (exact f32 math), then per-row sampled-softmax loss.
// One block per 16 rows; 4 waves, each wave owns one 16x16 output tile.
// ---------------------------------------------------------------------------
__global__ __launch_bounds__(128) void logits_loss_kernel(
    const float* __restrict__ item_table,
    const float* __restrict__ zero_bias,
    const int*   __restrict__ sampled_ids,
    const float* __restrict__ ut1,
    const int*   __restrict__ result_id,
    float*       __restrict__ row_loss)
{
    const int tid  = threadIdx.x;
    const int lane = tid & 31;
    const int wv   = tid >> 5;
    const int i0   = blockIdx.x * 16;  // row block
    const int j0   = wv * 16;          // col block (sampled id block)
    const int m    = lane & 15;        // M (for A/D) and N (for B/D)
    const int half = lane >> 4;
    const int koff = half * 2;

    __shared__ float sLog[16][S_SZ];

    // B[k][n] = samp_w[n][k] = item_table[sampled_ids[j0+n]][k]
    const int sid_l = sampled_ids[j0 + m];
    const float* aRow = ut1 + (i0 + m) * D_SZ + koff;
    const float* bRow = item_table + (long long)sid_l * D_SZ + koff;

    v8f c = {};
    #pragma unroll
    for (int kk = 0; kk < D_SZ; kk += 4) {
        v2f a  = *(const v2f*)(aRow + kk);  // A[m][kk+koff .. +1]
        v2f bv = *(const v2f*)(bRow + kk);  // B[kk+koff][n] pair
        c = __builtin_amdgcn_wmma_f32_16x16x4_f32(
                /*neg_a=*/false, a, /*neg_b=*/false, bv,
                /*c_mod=*/(short)0, c, /*reuse_a=*/false, /*reuse_b=*/false);
    }
    #pragma unroll
    for (int r = 0; r < 8; ++r)
        sLog[r + 8 * half][j0 + m] = c[r];   // D: row = r + 8*half, col = lane&15
    __syncthreads();

    // Per-row loss: 16 threads, one row each (fixed-order, deterministic).
    if (tid < 16) {
        const int g   = i0 + tid;
        const int res = result_id[g];
        const float* urow = ut1 + g * D_SZ;
        const float* wrow = item_table + (long long)res * D_SZ;
        float td = 0.0f;
        for (int k = 0; k < D_SZ; ++k) td += urow[k] * wrow[k];
        float tl = td + zero_bias[res] - logq_of(res);

        float mx = tl;
        for (int s = 0; s < S_SZ; ++s) {
            int sid = sampled_ids[s];
            float l = sLog[tid][s] + zero_bias[sid] - logq_of(sid);
            if (sid == res) l -= 1.0e9f;   // hit masking
            sLog[tid][s] = l;
            mx = fmaxf(mx, l);
        }
        float sum = expf(tl - mx);
        for (int s = 0; s < S_SZ; ++s) sum += expf(sLog[tid][s] - mx);
        row_loss[g] = (mx + logf(sum)) - tl;   // -log_softmax[:,0]
    }
}

// ---------------------------------------------------------------------------
// Kernel 3: deterministic mean over B row losses.
// ---------------------------------------------------------------------------
__global__ __launch_bounds__(256) void loss_reduce_kernel(
    const float* __restrict__ row_loss, float* __restrict__ loss_out)
{
    __shared__ float sdata[256];
    const int tid = threadIdx.x;
    float acc = 0.0f;
    for (int i = tid; i < B_SZ; i += 256) acc += row_loss[i];
    sdata[tid] = acc;
    __syncthreads();
    for (int s = 128; s > 0; s >>= 1) {
        if (tid < s) sdata[tid] += sdata[tid + s];
        __syncthreads();
    }
    if (tid == 0) loss_out[0] = sdata[0] / (float)B_SZ;
}

extern "C" void kernel_launch(void* const* d_in, const int* in_sizes, int n_in,
                              void* d_out, int out_size, void* d_ws, size_t ws_size,
                              hipStream_t stream) {
    const float* item_table  = (const float*)d_in[0];
    const float* pos_table   = (const float*)d_in[1];
    const float* att_W       = (const float*)d_in[2];
    const float* att_b       = (const float*)d_in[3];
    const float* prelu_alpha = (const float*)d_in[4];
    const float* zero_bias   = (const float*)d_in[5];
    const int*   items_id    = (const int*)d_in[6];
    const int*   position_id = (const int*)d_in[7];
    const int*   target_id   = (const int*)d_in[8];
    const int*   keys_length = (const int*)d_in[9];
    const int*   sampled_ids = (const int*)d_in[10];

    float* out = (float*)d_out;                       // [0..B): output, [B]: loss
    char*  ws  = (char*)d_ws;
    float* ut1       = (float*)ws;                                     // B*D f32
    int*   result_id = (int*)(ws + (size_t)B_SZ * D_SZ * sizeof(float));
    float* row_loss  = (float*)(ws + (size_t)B_SZ * D_SZ * sizeof(float)
                                   + (size_t)B_SZ * sizeof(int));

    attn_pool_kernel<<<B_SZ, 256, 0, stream>>>(
        item_table, pos_table, att_W, att_b, prelu_alpha,
        items_id, position_id, target_id, keys_length,
        out, ut1, result_id);

    logits_loss_kernel<<<B_SZ / 16, 128, 0, stream>>>(
        item_table, zero_bias, sampled_ids, ut1, result_id, row_loss);

    loss_reduce_kernel<<<1, 256, 0, stream>>>(row_loss, out + B_SZ);
}